// BboxTargetLayer_41901700939962
// MI455X (gfx1250) — compile-verified
//
#include <hip/hip_runtime.h>
#include <math.h>

// ---------------- constants from the reference ----------------
#define G_BOX      50
#define TOTAL_A    36864          // 64*64*9
#define NEG_OV_F   0.3f
#define POS_OV_F   0.7f
#define NUM_FG_I   128            // int(0.5*256)
#define RPN_BATCH_I 256

typedef unsigned int v4u __attribute__((ext_vector_type(4)));
typedef int          v8i __attribute__((ext_vector_type(8)));
typedef int          v4i __attribute__((ext_vector_type(4)));

// -------- TDM: stage one batch's GT table (50x5 f32 = 1000B) to LDS offset 0 --------
// D# group0: count=1, lds_addr=0, global_addr, type=2 ("image")
// D# group1: data_size=4B, tensor_dim0=250, tensor_dim1=1, tile=250x1, strides=250
__device__ __forceinline__ void tdm_load_gt(const float* gt, int b) {
    unsigned long long ga =
        (unsigned long long)(uintptr_t)(gt + (size_t)b * (G_BOX * 5));
    v4u g0;
    g0.x = 1u;                                         // count=1, is_restore=0
    g0.y = 0u;                                         // lds_addr = 0 (first shared var)
    g0.z = (unsigned)(ga & 0xffffffffull);             // global_addr[31:0]
    g0.w = (unsigned)((ga >> 32) & 0x01ffffffull) | (2u << 30); // addr[56:32] | type=2
    v8i g1;
    g1[0] = (int)(2u << 16);        // workgroup_mask=0, data_size=2 (4 bytes)
    g1[1] = (int)(250u << 16);      // tensor_dim0[15:0]=250
    g1[2] = (int)(1u << 16);        // tensor_dim0 hi=0, tensor_dim1[15:0]=1
    g1[3] = (int)(250u << 16);      // tensor_dim1 hi=0, tile_dim0=250
    g1[4] = 1;                      // tile_dim1=1, tile_dim2=0
    g1[5] = 250;                    // tensor_dim0_stride[31:0]
    g1[6] = (int)(250u << 16);      // stride0 hi=0, tensor_dim1_stride lo=250
    g1[7] = 0;
    v4i z4 = {0, 0, 0, 0};
    v8i z8 = {0, 0, 0, 0, 0, 0, 0, 0};
    __builtin_amdgcn_tensor_load_to_lds(g0, g1, z4, z4, z8, 0);  // 6-arg toolchain form
    __builtin_amdgcn_s_wait_tensorcnt(0);
    asm volatile("" ::: "memory");
}

// -------- async global->LDS copy (ASYNCcnt path), one B32 per lane --------
// lds_off: byte offset in workgroup LDS; base: uniform SGPR64 base; boff: per-lane byte offset
__device__ __forceinline__ void async_copy_b32(int lds_off, const float* base, int boff) {
    asm volatile("global_load_async_to_lds_b32 %0, %1, %2"
                 :
                 : "v"(lds_off), "v"(boff), "s"(base)
                 : "memory");
}
__device__ __forceinline__ void wait_async0() {
    asm volatile("s_wait_asynccnt 0x0" ::: "memory");
}
__device__ __forceinline__ void wait_async2() {
    asm volatile("s_wait_asynccnt 0x2" ::: "memory");
}

// float atomic max via int/uint ordering trick (no NaNs present)
__device__ __forceinline__ void atomicMaxF(float* addr, float v) {
    if (v >= 0.0f) atomicMax((int*)addr, __float_as_int(v));
    else           atomicMin((unsigned int*)addr, __float_as_uint(v));
}

// IoU with reference's gt_zero/a_zero overrides. contract(off) so K1 and K3
// produce bit-identical values (needed for the ov == gt_max equality test).
__device__ __forceinline__ float ov_one(float ax1, float ay1, float ax2, float ay2,
                                        float a_area, int a_zero, const float* gg) {
#pragma clang fp contract(off)
    float gx1 = gg[0], gy1 = gg[1], gx2 = gg[2], gy2 = gg[3];
    float gw = gx2 - gx1 + 1.0f;
    float gh = gy2 - gy1 + 1.0f;
    float g_area = gw * gh;
    float iw = fminf(ax2, gx2) - fmaxf(ax1, gx1) + 1.0f;
    float ih = fminf(ay2, gy2) - fmaxf(ay1, gy1) + 1.0f;
    float inter = fmaxf(iw, 0.0f) * fmaxf(ih, 0.0f);
    float ov = inter / (a_area + g_area - inter);
    if (gw == 1.0f && gh == 1.0f) ov = 0.0f;
    if (a_zero) ov = -1.0f;
    return ov;
}

// K0: init output: labels region = -1, targets region = 0
__global__ void k_init_out(float* out, int nlab, int ntot) {
    int idx = blockIdx.x * 256 + threadIdx.x;
    if (idx < ntot) out[idx] = (idx < nlab) ? -1.0f : 0.0f;
}

// K0b: init workspace reductions
__global__ void k_init_ws(float* gtmax, int* sumfg, int B) {
    int idx = blockIdx.x * 256 + threadIdx.x;
    if (idx < B * G_BOX) gtmax[idx] = -INFINITY;
    if (idx < B)         sumfg[idx] = 0;
}

// K1: per (b,i) max/argmax over GT; per (b,g) max over anchors (LDS then global atomic)
__global__ void k_overlap(const float* __restrict__ anchors, const float* __restrict__ gt,
                          float* __restrict__ max_ov, int* __restrict__ amax,
                          float* __restrict__ gtmax, int N) {
    __shared__ float smem[320];          // [0,250): GT via TDM ; [256,320): per-g max
    int b = blockIdx.y, tid = threadIdx.x;
    if (tid < 64) smem[256 + tid] = -INFINITY;
    if (tid < 32) tdm_load_gt(gt, b);    // wave 0 only issues + waits TENSORcnt
    __syncthreads();

    int i = blockIdx.x * 256 + tid;
    if (i < N) {
        float ax1 = anchors[i * 4 + 0], ay1 = anchors[i * 4 + 1];
        float ax2 = anchors[i * 4 + 2], ay2 = anchors[i * 4 + 3];
        float aw = ax2 - ax1 + 1.0f, ah = ay2 - ay1 + 1.0f;
        float a_area = aw * ah;
        int a_zero = (aw == 1.0f && ah == 1.0f);
        float best = -INFINITY; int barg = 0;
        for (int g = 0; g < G_BOX; ++g) {
            float ov = ov_one(ax1, ay1, ax2, ay2, a_area, a_zero, &smem[g * 5]);
            if (ov > best) { best = ov; barg = g; }   // first-occurrence argmax
            atomicMaxF(&smem[256 + g], ov);
        }
        max_ov[(size_t)b * N + i] = best;
        amax[(size_t)b * N + i]   = barg;
    }
    __syncthreads();
    if (tid < G_BOX) atomicMaxF(&gtmax[b * G_BOX + tid], smem[256 + tid]);
}

// K2: gt_max == 0 -> 1e-5 (reference fixup)
__global__ void k_fix_gtmax(float* gtmax, int n) {
    int idx = blockIdx.x * 256 + threadIdx.x;
    if (idx < n && gtmax[idx] == 0.0f) gtmax[idx] = 1e-5f;
}

// K3: label assignment (-1/0/1) + fg count per batch
__global__ void k_labels(const float* __restrict__ anchors, const float* __restrict__ gt,
                         const float* __restrict__ max_ov, const float* __restrict__ gtmax,
                         float* __restrict__ L, int* __restrict__ sumfg, int N) {
    __shared__ float smem[320];          // [0,250): GT via TDM ; [256,306): gt_max
    int b = blockIdx.y, tid = threadIdx.x;
    if (tid < G_BOX) smem[256 + tid] = gtmax[b * G_BOX + tid];
    if (tid < 32) tdm_load_gt(gt, b);
    __syncthreads();

    int i = blockIdx.x * 256 + tid;
    if (i >= N) return;
    float ax1 = anchors[i * 4 + 0], ay1 = anchors[i * 4 + 1];
    float ax2 = anchors[i * 4 + 2], ay2 = anchors[i * 4 + 3];
    float aw = ax2 - ax1 + 1.0f, ah = ay2 - ay1 + 1.0f;
    float a_area = aw * ah;
    int a_zero = (aw == 1.0f && ah == 1.0f);

    float mo = max_ov[(size_t)b * N + i];
    float lbl = -1.0f;
    if (mo < NEG_OV_F) lbl = 0.0f;
    bool keep = false;
    for (int g = 0; g < G_BOX; ++g) {
        float ov = ov_one(ax1, ay1, ax2, ay2, a_area, a_zero, &smem[g * 5]);
        if (ov == smem[256 + g]) keep = true;   // bit-exact: same helper as K1
    }
    if (keep) lbl = 1.0f;
    if (mo >= POS_OV_F) lbl = 1.0f;
    L[(size_t)b * N + i] = lbl;
    if (lbl == 1.0f) atomicAdd(&sumfg[b], 1);
}

// K4/K5: stable-rank subsampling with async double-buffered LDS tiles.
// rank(i) = #{j : match(j) && (r[j]<r[i] || (==, j<i))}
// mode 0: fg (match 1.0, thr = NUM_FG); mode 1: bg (match 0.0, thr = 256 - min(sum_fg,128))
// LDS layout (bytes): buf k at k*2048: L tile [0,1024), R tile [1024,2048)
__global__ void k_sample(const float* __restrict__ Lsrc, float* __restrict__ Ldst,
                         const float* __restrict__ rnd, const int* __restrict__ sumfg,
                         int mode, int N) {
    __shared__ float smem[1024];         // 2 x (256 L + 256 R)
    int b = blockIdx.y, tid = threadIdx.x;
    int i = blockIdx.x * 256 + tid;
    const float* Lb = Lsrc + (size_t)b * N;
    const float* Rb = rnd  + (size_t)b * N;
    float matchval = (mode == 0) ? 1.0f : 0.0f;
    int thr = (mode == 0) ? NUM_FG_I : (RPN_BATCH_I - min(sumfg[b], NUM_FG_I));

    float myl = (i < N) ? Lb[i] : -2.0f;
    float myr = (i < N) ? Rb[i] : 0.0f;
    bool active = (myl == matchval);
    int cnt = 0;
    int T = (N + 255) / 256;

    // every lane issues exactly 2 async ops per tile (clamped index; extra
    // entries are never read) so per-wave ASYNCcnt bookkeeping is uniform
    auto issue_tile = [&](int t, int buf) {
        int jc = min(t * 256 + tid, N - 1);
        async_copy_b32(buf * 2048 + tid * 4,        Lb, jc * 4);
        async_copy_b32(buf * 2048 + 1024 + tid * 4, Rb, jc * 4);
    };

    issue_tile(0, 0);
    if (T > 1) issue_tile(1, 1);
    for (int t = 0; t < T; ++t) {
        if (t + 1 < T) wait_async2();    // tile t landed (in-order), t+1 in flight
        else           wait_async0();    // drain
        __syncthreads();                 // tile t visible to all waves
        if (active) {
            int buf = t & 1;
            int j0 = t * 256;
            int lim = min(256, N - j0);
            const float* sL = &smem[buf * 512];
            const float* sR = &smem[buf * 512 + 256];
            for (int jj = 0; jj < lim; ++jj) {
                int jg = j0 + jj;
                if (sL[jj] == matchval &&
                    (sR[jj] < myr || (sR[jj] == myr && jg < i))) cnt++;
            }
        }
        __syncthreads();                 // all waves done reading buf (t&1)
        if (t + 2 < T) issue_tile(t + 2, t & 1);   // refill behind the barrier
    }

    if (i < N) {
        float outv = myl;
        if (active && cnt >= thr) outv = -1.0f;
        Ldst[(size_t)b * N + i] = outv;
    }
}

// K6: scatter labels + compute regression targets into the full grids
__global__ void k_finalize(const float* __restrict__ anchors, const float* __restrict__ gt,
                           const float* __restrict__ L, const int* __restrict__ amax,
                           const int* __restrict__ inds, float* __restrict__ out,
                           int N, int B) {
    __shared__ float smem[256];
    int b = blockIdx.y, tid = threadIdx.x;
    if (tid < 32) tdm_load_gt(gt, b);
    __syncthreads();

    int i = blockIdx.x * 256 + tid;
    if (i >= N) return;
    int t = inds[i];
    out[(size_t)b * TOTAL_A + t] = L[(size_t)b * N + i];

    float ax1 = anchors[i * 4 + 0], ay1 = anchors[i * 4 + 1];
    float ax2 = anchors[i * 4 + 2], ay2 = anchors[i * 4 + 3];
    float aw = ax2 - ax1 + 1.0f, ah = ay2 - ay1 + 1.0f;
    float ex_cx = ax1 + 0.5f * aw, ex_cy = ay1 + 0.5f * ah;

    int ag = amax[(size_t)b * N + i];
    float gx1 = smem[ag * 5 + 0], gy1 = smem[ag * 5 + 1];
    float gx2 = smem[ag * 5 + 2], gy2 = smem[ag * 5 + 3];
    float tw = gx2 - gx1 + 1.0f, th = gy2 - gy1 + 1.0f;
    float tcx = gx1 + 0.5f * tw, tcy = gy1 + 0.5f * th;

    float* tgt = out + (size_t)B * TOTAL_A + ((size_t)b * TOTAL_A + t) * 4;
    tgt[0] = (tcx - ex_cx) / aw;
    tgt[1] = (tcy - ex_cy) / ah;
    tgt[2] = logf(tw / aw);
    tgt[3] = logf(th / ah);
}

extern "C" void kernel_launch(void* const* d_in, const int* in_sizes, int n_in,
                              void* d_out, int out_size, void* d_ws, size_t ws_size,
                              hipStream_t stream) {
    const float* gt      = (const float*)d_in[0];   // (B,50,5) f32
    const float* anchors = (const float*)d_in[1];   // (N,4)  f32
    const int*   inds    = (const int*)d_in[2];     // (N,)   i32
    const float* rfg     = (const float*)d_in[3];   // (B,N)  f32
    const float* rbg     = (const float*)d_in[4];   // (B,N)  f32
    int B = in_sizes[0] / (G_BOX * 5);
    int N = in_sizes[1] / 4;
    float* out = (float*)d_out;

    // workspace carve-up (256B aligned)
    char* w = (char*)d_ws;
    size_t off = 0;
    auto carve = [&](size_t bytes) {
        void* p = w + off;
        off = (off + bytes + 255) & ~(size_t)255;
        return p;
    };
    float* max_ov = (float*)carve((size_t)B * N * sizeof(float));
    int*   amax   = (int*)  carve((size_t)B * N * sizeof(int));
    float* L      = (float*)carve((size_t)B * N * sizeof(float));
    float* L2     = (float*)carve((size_t)B * N * sizeof(float));
    float* gtmax  = (float*)carve((size_t)B * G_BOX * sizeof(float));
    int*   sumfg  = (int*)  carve((size_t)B * sizeof(int));
    (void)ws_size; (void)n_in; (void)out_size;

    int ntot = B * TOTAL_A * 5;
    k_init_out<<<(ntot + 255) / 256, 256, 0, stream>>>(out, B * TOTAL_A, ntot);
    k_init_ws<<<(B * G_BOX + 255) / 256, 256, 0, stream>>>(gtmax, sumfg, B);

    dim3 grid((N + 255) / 256, B);
    k_overlap<<<grid, 256, 0, stream>>>(anchors, gt, max_ov, amax, gtmax, N);
    k_fix_gtmax<<<(B * G_BOX + 255) / 256, 256, 0, stream>>>(gtmax, B * G_BOX);
    k_labels<<<grid, 256, 0, stream>>>(anchors, gt, max_ov, gtmax, L, sumfg, N);
    k_sample<<<grid, 256, 0, stream>>>(L, L2, rfg, sumfg, 0, N);   // fg
    k_sample<<<grid, 256, 0, stream>>>(L2, L, rbg, sumfg, 1, N);   // bg
    k_finalize<<<grid, 256, 0, stream>>>(anchors, gt, L, amax, inds, out, N, B);
}